// MSBEGCL_Encoder_78030965833844
// MI455X (gfx1250) — compile-verified
//
#include <hip/hip_runtime.h>
#include <math.h>

#define USER_NUM_  100000
#define ITEM_NUM_  50000
#define N_NODES_   150000
#define NNZ_       4800000
#define EMB_       64
#define EPS_       0.2f
#define NORM_EPS_  1e-12f
#define NTILES_    (N_NODES_ / 16)   // 9375 tiles of 16 rows

typedef __attribute__((ext_vector_type(2))) float v2f;
typedef __attribute__((ext_vector_type(8))) float v8f;

// ---------------------------------------------------------------------------
// ego = concat(user_emb, item_emb)   (float4-vectorized copy)
// ---------------------------------------------------------------------------
__global__ void concat_kernel(const float* __restrict__ u,
                              const float* __restrict__ it,
                              float* __restrict__ ego) {
    size_t i = (size_t)blockIdx.x * blockDim.x + threadIdx.x;
    const size_t userQuads  = (size_t)USER_NUM_ * EMB_ / 4;
    const size_t totalQuads = (size_t)N_NODES_ * EMB_ / 4;
    if (i >= totalQuads) return;
    float4 v = (i < userQuads) ? ((const float4*)u)[i]
                               : ((const float4*)it)[i - userQuads];
    ((float4*)ego)[i] = v;
}

__global__ void zero_kernel(float4* __restrict__ p, size_t nQuads) {
    size_t i = (size_t)blockIdx.x * blockDim.x + threadIdx.x;
    if (i < nQuads) p[i] = make_float4(0.f, 0.f, 0.f, 0.f);
}

// ---------------------------------------------------------------------------
// COO SpMM: y[r,:] += v * x[c,:]   one wave per nonzero, float2 per lane,
// native f32 global atomics for the scatter-add.
// ---------------------------------------------------------------------------
__global__ void spmm_kernel(const int* __restrict__ rows,
                            const int* __restrict__ cols,
                            const float* __restrict__ vals,
                            const float* __restrict__ x,
                            float* __restrict__ y) {
    const int lane   = threadIdx.x & 31;
    const int wave   = (int)((blockIdx.x * blockDim.x + threadIdx.x) >> 5);
    const int stride = (int)((gridDim.x * blockDim.x) >> 5);
    for (int e = wave; e < NNZ_; e += stride) {
        const int   r = rows[e];          // wave-uniform loads (broadcast)
        const int   c = cols[e];
        const float v = vals[e];
        const float2 xv = *(const float2*)(x + (size_t)c * EMB_ + lane * 2);
        float* yp = y + (size_t)r * EMB_ + lane * 2;
        unsafeAtomicAdd(yp,     v * xv.x);   // global_atomic_add_f32
        unsafeAtomicAdd(yp + 1, v * xv.y);
    }
}

// ---------------------------------------------------------------------------
// Perturb + mean accumulate.  One wave handles 16 rows (a 16x64 noise tile).
// Row L2-norms come from the diagonal of G = R * R^T computed with
// V_WMMA_F32_16X16X4_F32 accumulated over K in 16 steps.  For the f32 16x4
// A layout (lanes 0-15: K=0..1 in v[0..1]; lanes 16-31: K=2..3) the matching
// B (4x16) layout of R^T occupies identical registers, so A==B per step.
// ---------------------------------------------------------------------------
__device__ __forceinline__ float sgnf(float x) {
    return (x > 0.f) ? 1.f : ((x < 0.f) ? -1.f : 0.f);
}

template <int FIRST>
__global__ void perturb_kernel(float* __restrict__ ego,
                               const float* __restrict__ noise,
                               float* __restrict__ out) {
    const int lane = threadIdx.x & 31;
    const int wave = (int)((blockIdx.x * blockDim.x + threadIdx.x) >> 5);
    if (wave >= NTILES_) return;          // wave-uniform: EXEC stays all-ones
    const int rowBase = wave * 16;
    const int laneLo  = lane & 15;
    const int laneHi  = lane >> 4;

    // Gram accumulation: acc[m][n] = dot(noise[rowBase+m,:], noise[rowBase+n,:])
    const float* nb = noise + (size_t)(rowBase + laneLo) * EMB_ + laneHi * 2;
    v8f acc = {};
#pragma unroll
    for (int t = 0; t < 16; ++t) {
        const float2 d = *(const float2*)(nb + 4 * t);
        v2f ab;
        ab.x = d.x;
        ab.y = d.y;
        acc = __builtin_amdgcn_wmma_f32_16x16x4_f32(
            /*neg_a=*/false, ab, /*neg_b=*/false, ab,
            /*c_mod=*/(short)0, acc, /*reuse_a=*/false, /*reuse_b=*/false);
    }

    // Extract diag(G): m<8 -> (vgpr m, lane m); m>=8 -> (vgpr m-8, lane 24+(m-8))
    float norms[16];
#pragma unroll
    for (int m = 0; m < 8; ++m) {
        norms[m]     = __shfl(acc[m], m,      32);
        norms[m + 8] = __shfl(acc[m], 24 + m, 32);
    }

    const float inv3 = 1.0f / 3.0f;
#pragma unroll
    for (int m = 0; m < 16; ++m) {
        float nrm = sqrtf(norms[m]);
        nrm = fmaxf(nrm, NORM_EPS_);
        const float scale = EPS_ / nrm;
        const size_t idx = (size_t)(rowBase + m) * EMB_ + lane * 2;
        const float2 s = *(const float2*)(ego + idx);
        const float2 r = *(const float2*)(noise + idx);
        const float ex = s.x + sgnf(s.x) * r.x * scale;
        const float ey = s.y + sgnf(s.y) * r.y * scale;
        *(float2*)(ego + idx) = make_float2(ex, ey);   // feeds next layer's SpMM
        float2 o;
        if (FIRST) {
            o = make_float2(ex * inv3, ey * inv3);     // overwrite poisoned d_out
        } else {
            o = *(const float2*)(out + idx);
            o.x += ex * inv3;
            o.y += ey * inv3;
        }
        *(float2*)(out + idx) = o;
    }
}

// ---------------------------------------------------------------------------
// Driver
// ---------------------------------------------------------------------------
extern "C" void kernel_launch(void* const* d_in, const int* in_sizes, int n_in,
                              void* d_out, int out_size, void* d_ws, size_t ws_size,
                              hipStream_t stream) {
    const float* user_emb = (const float*)d_in[0];
    const float* item_emb = (const float*)d_in[1];
    const int*   adj_rows = (const int*)d_in[2];
    const int*   adj_cols = (const int*)d_in[3];
    const float* adj_vals = (const float*)d_in[4];
    const float* noise    = (const float*)d_in[5];
    float* out = (float*)d_out;

    const size_t embFloats = (size_t)N_NODES_ * EMB_;       // 9.6M floats
    const size_t embQuads  = embFloats / 4;                 // 2.4M float4
    float* egoA = (float*)d_ws;                             // 38.4 MB
    float* egoB = egoA + embFloats;                         // 38.4 MB

    const int COPY_BLOCKS = (int)((embQuads + 255) / 256);  // 9375
    const int SPMM_BLOCKS = 2048;                           // 16384 waves, grid-stride
    const int PERT_BLOCKS = (NTILES_ * 32 + 255) / 256;     // 1172

    // Layer-0 input: concat(user_emb, item_emb)
    concat_kernel<<<COPY_BLOCKS, 256, 0, stream>>>(user_emb, item_emb, egoA);

    float* src = egoA;
    float* dst = egoB;
    for (int k = 0; k < 3; ++k) {
        zero_kernel<<<COPY_BLOCKS, 256, 0, stream>>>((float4*)dst, embQuads);
        spmm_kernel<<<SPMM_BLOCKS, 256, 0, stream>>>(adj_rows, adj_cols, adj_vals, src, dst);
        const float* nk = noise + (size_t)k * N_NODES_ * EMB_;
        if (k == 0) perturb_kernel<1><<<PERT_BLOCKS, 256, 0, stream>>>(dst, nk, out);
        else        perturb_kernel<0><<<PERT_BLOCKS, 256, 0, stream>>>(dst, nk, out);
        float* t = src; src = dst; dst = t;                 // ping-pong
    }
}